// V9StyleCodebook_16587163697601
// MI455X (gfx1250) — compile-verified
//
#include <hip/hip_runtime.h>

typedef float v2f __attribute__((ext_vector_type(2)));
typedef float v8f __attribute__((ext_vector_type(8)));
typedef int   v4i __attribute__((vector_size(16)));

#define B_ROWS   262144
#define D_DIM    256
#define K_CODES  512
#define ROW_PAD  260          // 260 dwords: 16 B-frag lanes land on distinct banks
#define CHUNK    64           // codes staged in LDS per group
#define NGROUPS  (K_CODES / CHUNK)   // 8
#define WG_ROWS  128          // 8 waves * 16 rows
#define KSTEPS   (D_DIM / 4)  // 64 wmma k-steps

#if defined(__has_builtin)
#if __has_builtin(__builtin_amdgcn_global_load_async_to_lds_b128) && \
    __has_builtin(__builtin_amdgcn_s_wait_asynccnt)
#define USE_ASYNC_LDS 1
#endif
#endif
#ifndef USE_ASYNC_LDS
#define USE_ASYNC_LDS 0
#endif

typedef __attribute__((address_space(1))) v4i* gptr_v4i;   // global int4*
typedef __attribute__((address_space(3))) v4i* lptr_v4i;   // LDS int4*

// ---------------------------------------------------------------------------
// Kernel 1: e_sq[k] = sum_d embed[k][d]^2   (tiny: 512 rows)
// ---------------------------------------------------------------------------
__global__ void vq_esq_kernel(const float* __restrict__ embed,
                              float* __restrict__ esq) {
    int k = blockIdx.x * blockDim.x + threadIdx.x;
    if (k < K_CODES) {
        const float* row = embed + (size_t)k * D_DIM;
        float s = 0.f;
        #pragma unroll 8
        for (int d = 0; d < D_DIM; ++d) s = fmaf(row[d], row[d], s);
        esq[k] = s;
    }
}

// ---------------------------------------------------------------------------
// Kernel 2: main VQ kernel. 256 threads = 8 waves; each wave owns 16 rows.
// ---------------------------------------------------------------------------
__global__ __launch_bounds__(256) void vq_main_kernel(
    const float* __restrict__ z,
    const float* __restrict__ embed,
    const float* __restrict__ esq,
    float* __restrict__ qout,
    int*   __restrict__ codes,
    float* __restrict__ commit)
{
#if USE_ASYNC_LDS
    __shared__ float lds_e[2 * CHUNK * ROW_PAD];   // 133,120 B (double buffer)
#else
    __shared__ float lds_e[CHUNK * ROW_PAD];       //  66,560 B
#endif
    __shared__ float lds_esq[K_CODES];             //   2,048 B

    const int tid  = threadIdx.x;
    const int wave = tid >> 5;
    const int lane = tid & 31;
    const int ln   = lane & 15;      // M (A) / N (B) index
    const int h    = lane >> 4;      // K-half selector
    const int row0 = blockIdx.x * WG_ROWS + wave * 16;

    // stage e_sq once
    lds_esq[tid]       = esq[tid];
    lds_esq[tid + 256] = esq[tid + 256];

    // ---- preload A fragments for all of D (z read exactly once from HBM) ---
    // A 16x4 f32 layout: lane ln holds row M=ln; vgpr0/1 = K = 2*h + {0,1}
    v2f a[KSTEPS];
    const float* zrow = z + (size_t)(row0 + ln) * D_DIM + 2 * h;
    float zsq = 0.f;
    #pragma unroll
    for (int kk = 0; kk < KSTEPS; ++kk) {
        v2f t = *(const v2f*)(zrow + kk * 4);
        a[kk] = t;
        zsq = fmaf(t.x, t.x, zsq);
        zsq = fmaf(t.y, t.y, zsq);
    }

    float minval[8];
    int   minidx[8];
    #pragma unroll
    for (int v = 0; v < 8; ++v) { minval[v] = 3.402823466e+38f; minidx[v] = 0; }

#if USE_ASYNC_LDS
    // ---- async double-buffered staging: copies of group g+1 overlap the ----
    // ---- 256 WMMAs/wave of group g; no staging VGPRs, tracked by ASYNCcnt.  ----
    {
        // issue group 0 into buffer 0
        const float* src0 = embed;                 // g = 0
        #pragma unroll
        for (int i = 0; i < 16; ++i) {
            int f  = tid + i * 256;
            int r  = f >> 6;
            int c4 = f & 63;
            __builtin_amdgcn_global_load_async_to_lds_b128(
                (gptr_v4i)(src0 + (size_t)f * 4),
                (lptr_v4i)&lds_e[r * ROW_PAD + c4 * 4], 0, 0);
        }
    }
    for (int g = 0; g < NGROUPS; ++g) {
        __builtin_amdgcn_s_wait_asynccnt(0);   // my copies of group g landed
        __syncthreads();                       // everyone's copies landed AND
                                               // everyone finished group g-1
        if (g + 1 < NGROUPS) {                 // prefetch g+1 over compute(g)
            const float* srcn = embed + (size_t)(g + 1) * CHUNK * D_DIM;
            float* bufn = lds_e + ((g + 1) & 1) * (CHUNK * ROW_PAD);
            #pragma unroll
            for (int i = 0; i < 16; ++i) {
                int f  = tid + i * 256;
                int r  = f >> 6;
                int c4 = f & 63;
                __builtin_amdgcn_global_load_async_to_lds_b128(
                    (gptr_v4i)(srcn + (size_t)f * 4),
                    (lptr_v4i)&bufn[r * ROW_PAD + c4 * 4], 0, 0);
            }
        }
        const float* buf = lds_e + (g & 1) * (CHUNK * ROW_PAD);
#else
    for (int g = 0; g < NGROUPS; ++g) {
        __syncthreads();
        {   // synchronous cooperative stage of 64 codes x 256 f32
            const float4* src = (const float4*)(embed + (size_t)g * CHUNK * D_DIM);
            #pragma unroll
            for (int i = 0; i < 16; ++i) {
                int f  = tid + i * 256;
                int r  = f >> 6;
                int c4 = f & 63;
                float4 val = src[f];
                *(float4*)&lds_e[r * ROW_PAD + c4 * 4] = val;
            }
        }
        __syncthreads();
        const float* buf = lds_e;
#endif

        for (int s = 0; s < 4; ++s) {       // four 16-code sub-tiles
            v8f acc = {};
            // B 4x16 f32 layout: lane ln holds col N=ln; vgpr0/1 = K = 2*h+{0,1}
            const float* bbase = &buf[(s * 16 + ln) * ROW_PAD + 2 * h];
            #pragma unroll
            for (int kk = 0; kk < KSTEPS; ++kk) {
                v2f b = *(const v2f*)(bbase + kk * 4);
                acc = __builtin_amdgcn_wmma_f32_16x16x4_f32(
                    false, a[kk], false, b, (short)0, acc, false, false);
            }
            const int   code = g * CHUNK + s * 16 + ln;
            const float es   = lds_esq[code];
            // C layout: vgpr v, lanes 0-15: row v; lanes 16-31: row v+8; col=ln
            #pragma unroll
            for (int v = 0; v < 8; ++v) {
                float sc = fmaf(-2.f, acc[v], es);  // ||z||^2 dropped (row const)
                if (sc < minval[v]) { minval[v] = sc; minidx[v] = code; }
            }
        }
    }

    // ---- cross-lane argmin butterfly within each 16-lane half ---------------
    float rval[8]; int ridx[8];
    #pragma unroll
    for (int v = 0; v < 8; ++v) {
        float val = minval[v]; int idx = minidx[v];
        #pragma unroll
        for (int off = 1; off < 16; off <<= 1) {
            float ov = __shfl_xor(val, off, 32);
            int   oi = __shfl_xor(idx, off, 32);
            if (ov < val || (ov == val && oi < idx)) { val = ov; idx = oi; }
        }
        rval[v] = val; ridx[v] = idx;
    }

    // ---- codes (rows 0-7 from lane 0, rows 8-15 from lane 16) ---------------
    if (lane == 0) {
        #pragma unroll
        for (int v = 0; v < 8; ++v) codes[row0 + v] = ridx[v];
    } else if (lane == 16) {
        #pragma unroll
        for (int v = 0; v < 8; ++v) codes[row0 + 8 + v] = ridx[v];
    }

    // ---- commitment loss partial: sum_r (||z_r||^2 + minscore_r) ------------
    float msum = 0.f;
    #pragma unroll
    for (int v = 0; v < 8; ++v) msum += rval[v];
    msum += __shfl_xor(msum, 16, 32);        // low-rows + high-rows
    float zs = zsq;
    #pragma unroll
    for (int off = 1; off < 32; off <<= 1) zs += __shfl_xor(zs, off, 32);
    if (lane == 0) {
        float p = (zs + msum) * (0.25f / ((float)B_ROWS * (float)D_DIM));
        atomicAdd(commit, p);
    }

    // ---- quantized = embed[code]  (straight-through forward value) ----------
    #pragma unroll
    for (int r = 0; r < 16; ++r) {
        int c = __shfl(ridx[r & 7], (r < 8) ? 0 : 16, 32);
        const float4* src = (const float4*)(embed + (size_t)c * D_DIM);
        float4*       dst = (float4*)(qout + (size_t)(row0 + r) * D_DIM);
        dst[lane]      = src[lane];
        dst[lane + 32] = src[lane + 32];
    }
}

// ---------------------------------------------------------------------------
extern "C" void kernel_launch(void* const* d_in, const int* in_sizes, int n_in,
                              void* d_out, int out_size, void* d_ws, size_t ws_size,
                              hipStream_t stream) {
    const float* z     = (const float*)d_in[0];
    const float* embed = (const float*)d_in[1];

    const size_t BD = (size_t)B_ROWS * D_DIM;
    float* qout   = (float*)d_out;
    int*   codes  = (int*)((float*)d_out + BD);
    float* commit = (float*)d_out + BD + B_ROWS;
    float* esq    = (float*)d_ws;

    (void)hipMemsetAsync(commit, 0, sizeof(float), stream);
    vq_esq_kernel<<<2, 256, 0, stream>>>(embed, esq);
    vq_main_kernel<<<B_ROWS / WG_ROWS, 256, 0, stream>>>(
        z, embed, esq, qout, codes, commit);
}